// ModDGCNN_68564857914112
// MI455X (gfx1250) — compile-verified
//
#include <hip/hip_runtime.h>

typedef __attribute__((ext_vector_type(2))) float v2f;
typedef __attribute__((ext_vector_type(8))) float v8f;

#define NVAL 1024
#define NEGV (-1e30f)

__device__ __forceinline__ v8f wmma_f32(v2f a, v2f b, v8f c) {
  // V_WMMA_F32_16X16X4_F32: D = A(16x4) * B(4x16) + C(16x16), fp32, wave32
  return __builtin_amdgcn_wmma_f32_16x16x4_f32(false, a, false, b, (short)0, c, false, false);
}

// ---------------------------------------------------------------------------
// Blockwise KNN: per (batch, block i of 32 points) compute 32x224 distance
// matrix with WMMA (Gram matrix 2*q.c), then iterative top-64 per query.
// ---------------------------------------------------------------------------
__global__ __launch_bounds__(256) void knn_kernel(const float* __restrict__ x, long xbs,
                                                  int C, int* __restrict__ idx) {
  __shared__ float dmat[32][224];   // 28 KB
  __shared__ float cc[224];         // candidate squared norms
  __shared__ int   ccol[224];       // clipped candidate global columns
  const int N = NVAL;
  const int bi = blockIdx.x;        // block row i (0..31)
  const int b  = blockIdx.y;
  const float* xb = x + (long)b * xbs;
  const int tid = threadIdx.x;

  // candidate columns (clipped like the reference gather) + squared norms
  for (int j = tid; j < 224; j += 256) {
    int wb  = bi - 3 + (j >> 5);
    int wbc = wb < 0 ? 0 : (wb > 31 ? 31 : wb);
    int cn  = (wbc << 5) + (j & 31);
    ccol[j] = cn;
    float s = 0.f;
    for (int c = 0; c < C; ++c) { float t = xb[(long)c * N + cn]; s += t * t; }
    cc[j] = s;
  }
  __syncthreads();

  const int lane = tid & 31;
  const int wv   = tid >> 5;        // wave id 0..7
  // 2 x 14 = 28 output tiles of 16x16, round-robin over 8 waves
  for (int tile = wv; tile < 28; tile += 8) {
    int m0 = (tile / 14) << 4;
    int n0 = (tile % 14) << 4;
    int qn = (bi << 5) + m0 + (lane & 15);   // query column (A: lane = M)
    int cn = ccol[n0 + (lane & 15)];         // candidate column (B: lane = N)
    int kh = (lane >> 4) << 1;               // K half: lanes 16-31 hold K=2,3
    v8f acc = {};
    for (int c = 0; c < C; c += 4) {
      const float* x0 = xb + (long)(c + kh) * N;
      v2f a  = { x0[qn], x0[N + qn] };       // A[m][kh], A[m][kh+1]
      v2f bt = { x0[cn], x0[N + cn] };       // B[kh][n], B[kh+1][n]
      acc = wmma_f32(a, bt, acc);
    }
#pragma unroll
    for (int r = 0; r < 8; ++r) {
      int m = m0 + r + ((lane >> 4) << 3);   // C/D layout: lanes 16-31 -> M+8
      int j = n0 + (lane & 15);
      int wb = bi - 3 + (j >> 5);
      dmat[m][j] = (wb < 0 || wb > 31) ? NEGV
                 : (2.f * acc[r] - cc[96 + m] - cc[j]);  // cc[96+m] == |q|^2
    }
  }
  __syncthreads();

  // top-64 per query row: 4 queries per wave, extract-max 64 times
  for (int q = wv; q < 32; q += 8) {
    float loc[7];
#pragma unroll
    for (int t = 0; t < 7; ++t) loc[t] = dmat[q][(t << 5) + lane];
    for (int it = 0; it < 64; ++it) {
      float best = -3.0e38f; int bj = 0x7fffffff;
#pragma unroll
      for (int t = 0; t < 7; ++t) {
        float vv = loc[t]; int jj = (t << 5) + lane;
        if (vv > best || (vv == best && jj < bj)) { best = vv; bj = jj; }
      }
#pragma unroll
      for (int off = 16; off > 0; off >>= 1) {
        float ov = __shfl_xor(best, off);
        int   oj = __shfl_xor(bj, off);
        if (ov > best || (ov == best && oj < bj)) { best = ov; bj = oj; }
      }
      if (lane == 0)
        idx[((long)b * NVAL + (bi << 5) + q) * 64 + it] = ((bi - 3) << 5) + bj;
      if (lane == (bj & 31)) loc[bj >> 5] = -3.0e38f;   // remove winner
    }
  }
}

// ---------------------------------------------------------------------------
// u = (w2-w1) @ x, v = w1 @ x   (w is O x 2C, split [w1|w2]).
// 256-thread blocks = 8 waves; each wave owns a 16(o) x 64(n) strip (4 tiles),
// reusing the A (weight) fragment across 4 WMMAs per matrix per K-step.
// ---------------------------------------------------------------------------
__global__ __launch_bounds__(256) void uv_gemm_kernel(const float* __restrict__ w,
    const float* __restrict__ x, float* __restrict__ u, float* __restrict__ v,
    int C, int O, long xbs) {
  const int N = NVAL;
  const int lane = threadIdx.x & 31;
  const int wv   = threadIdx.x >> 5;
  const int n0 = (blockIdx.x << 9) + (wv << 6);   // 512 cols/block, 64/wave
  const int o0 = blockIdx.y << 4;
  const int b  = blockIdx.z;
  const float* xb = x + (long)b * xbs;
  const int mrow = o0 + (lane & 15);
  const int kh   = (lane >> 4) << 1;
  const int nc   = n0 + (lane & 15);
  v8f aU0 = {}, aU1 = {}, aU2 = {}, aU3 = {};
  v8f aV0 = {}, aV1 = {}, aV2 = {}, aV3 = {};
  for (int c = 0; c < C; c += 4) {
    const float* wr = w + (long)mrow * (2 * C) + c + kh;
    float w1x = wr[0], w1y = wr[1];
    v2f fV = { w1x, w1y };                        // w1
    v2f fU = { wr[C] - w1x, wr[C + 1] - w1y };    // w2 - w1
    const float* xc = xb + (long)(c + kh) * N + nc;
    v2f b0 = { xc[0],  xc[N]      };
    v2f b1 = { xc[16], xc[N + 16] };
    v2f b2 = { xc[32], xc[N + 32] };
    v2f b3 = { xc[48], xc[N + 48] };
    aV0 = wmma_f32(fV, b0, aV0);  aU0 = wmma_f32(fU, b0, aU0);
    aV1 = wmma_f32(fV, b1, aV1);  aU1 = wmma_f32(fU, b1, aU1);
    aV2 = wmma_f32(fV, b2, aV2);  aU2 = wmma_f32(fU, b2, aU2);
    aV3 = wmma_f32(fV, b3, aV3);  aU3 = wmma_f32(fU, b3, aU3);
  }
  const int mo = o0 + ((lane >> 4) << 3);
  const int no = n0 + (lane & 15);
#pragma unroll
  for (int r = 0; r < 8; ++r) {
    long base = ((long)b * O + mo + r) * N + no;
    u[base]      = aU0[r];  v[base]      = aV0[r];
    u[base + 16] = aU1[r];  v[base + 16] = aV1[r];
    u[base + 32] = aU2[r];  v[base + 32] = aV2[r];
    u[base + 48] = aU3[r];  v[base + 48] = aV3[r];
  }
}

// ---------------------------------------------------------------------------
// Per (b, n-tile of 8): gather v at the 64 neighbor indices, accumulate
// sum / sum^2 / max; pre = u + max_k v (BN+LeakyReLU are monotonic, so max
// commutes).  BN stats partials written per block (deterministic reduction).
// global_prefetch_b8 pulls the next index row while gathering.
// ---------------------------------------------------------------------------
__global__ void gather_stats_kernel(const float* __restrict__ u, const float* __restrict__ v,
    const int* __restrict__ idx, float* __restrict__ pre, float* __restrict__ part, int O) {
  const int N = NVAL;
  __shared__ int sidx[64];
  const int o = threadIdx.x;
  const int b = blockIdx.y;
  const int p = blockIdx.y * gridDim.x + blockIdx.x;
  const float* vrow = v + ((long)b * O + o) * N;
  const float* urow = u + ((long)b * O + o) * N;
  float* prow = pre + ((long)b * O + o) * N;
  float psy = 0.f, psy2 = 0.f;
  for (int nn = 0; nn < 8; ++nn) {
    int n = (blockIdx.x << 3) + nn;
    __syncthreads();
    if (o < 64) sidx[o] = idx[((long)b * N + n) * 64 + o];
    __syncthreads();
    if (nn < 7 && o < 2)   // prefetch next 256-byte index row (global_prefetch_b8)
      __builtin_prefetch(idx + ((long)b * N + n + 1) * 64 + (o << 5), 0, 1);
    float s = 0.f, s2 = 0.f, m = -3.0e38f;
#pragma unroll 8
    for (int k = 0; k < 64; ++k) {
      float t = vrow[sidx[k]];
      s += t; s2 += t * t; m = fmaxf(m, t);
    }
    float uv = urow[n];
    prow[n] = uv + m;
    psy  += s + 64.f * uv;                       // sum_k (v_g + u)
    psy2 += s2 + 2.f * uv * s + 64.f * uv * uv;  // sum_k (v_g + u)^2
  }
  part[(long)p * (2 * O) + o]     = psy;
  part[(long)p * (2 * O) + O + o] = psy2;
}

__global__ void stats_reduce_kernel(const float* __restrict__ part, float* __restrict__ stats,
                                    int O, int npart) {
  int o = threadIdx.x;
  if (o >= 2 * O) return;
  float s = 0.f;
  for (int p = 0; p < npart; ++p) s += part[(long)p * (2 * O) + o];
  stats[o] = s;
}

__global__ void bn_act_kernel(const float* __restrict__ pre, const float* __restrict__ stats,
    const float* __restrict__ g, const float* __restrict__ bb, float* __restrict__ out,
    int O, long obs, float inv) {
  const int N = NVAL;
  long i = (long)blockIdx.x * blockDim.x + threadIdx.x;   // over (b, o, n)
  int n = (int)(i % N);
  long t = i / N;
  int o = (int)(t % O);
  int b = (int)(t / O);
  float mu  = stats[o] * inv;
  float var = stats[O + o] * inv - mu * mu;
  float y = (pre[i] - mu) * rsqrtf(var + 1e-5f) * g[o] + bb[o];
  y = y >= 0.f ? y : 0.2f * y;
  out[(long)b * obs + (long)o * N + n] = y;
}

// ---------------------------------------------------------------------------
// Plain fp32 WMMA GEMM: out[b,o,n] = sum_c w[o,c] * x[b,c,n]
// Each wave: 16(o) x 64(n) strip, A fragment reused across 4 WMMAs per step.
// ---------------------------------------------------------------------------
__global__ __launch_bounds__(256) void gemm_kernel(const float* __restrict__ w,
    const float* __restrict__ x, float* __restrict__ out, int C, long xbs, long obs) {
  const int N = NVAL;
  const int lane = threadIdx.x & 31;
  const int wv   = threadIdx.x >> 5;
  const int n0 = (blockIdx.x << 9) + (wv << 6);
  const int o0 = blockIdx.y << 4;
  const int b  = blockIdx.z;
  const float* xb = x + (long)b * xbs;
  const int mrow = o0 + (lane & 15);
  const int kh   = (lane >> 4) << 1;
  const int nc   = n0 + (lane & 15);
  v8f a0 = {}, a1 = {}, a2 = {}, a3 = {};
  for (int c = 0; c < C; c += 4) {
    const float* wr = w + (long)mrow * C + c + kh;
    v2f fA = { wr[0], wr[1] };
    const float* xc = xb + (long)(c + kh) * N + nc;
    v2f b0 = { xc[0],  xc[N]      };
    v2f b1 = { xc[16], xc[N + 16] };
    v2f b2 = { xc[32], xc[N + 32] };
    v2f b3 = { xc[48], xc[N + 48] };
    a0 = wmma_f32(fA, b0, a0);
    a1 = wmma_f32(fA, b1, a1);
    a2 = wmma_f32(fA, b2, a2);
    a3 = wmma_f32(fA, b3, a3);
  }
  const int mo = o0 + ((lane >> 4) << 3);
  const int no = n0 + (lane & 15);
#pragma unroll
  for (int r = 0; r < 8; ++r) {
    long base = (long)b * obs + (long)(mo + r) * N + no;
    out[base]      = a0[r];
    out[base + 16] = a1[r];
    out[base + 32] = a2[r];
    out[base + 48] = a3[r];
  }
}

extern "C" void kernel_launch(void* const* d_in, const int* in_sizes, int n_in,
                              void* d_out, int out_size, void* d_ws, size_t ws_size,
                              hipStream_t stream) {
  (void)in_sizes; (void)n_in; (void)out_size; (void)ws_size;
  const int N = NVAL, B = 2;
  const float* x0in  = (const float*)d_in[0];
  const float* W[4]  = {(const float*)d_in[1], (const float*)d_in[4],
                        (const float*)d_in[7], (const float*)d_in[10]};
  const float* G[4]  = {(const float*)d_in[2], (const float*)d_in[5],
                        (const float*)d_in[8], (const float*)d_in[11]};
  const float* Bi[4] = {(const float*)d_in[3], (const float*)d_in[6],
                        (const float*)d_in[9], (const float*)d_in[12]};
  const float* wlast = (const float*)d_in[13];
  const int Cs[4]   = {256, 128, 128, 64};
  const int Os[4]   = {128, 128, 64, 64};
  const int coff[4] = {0, 128, 256, 320};   // channel offsets inside xcon (384 ch)

  float* ws    = (float*)d_ws;
  float* xcon  = ws;                         // 2*384*1024
  int*   idx   = (int*)(ws + 786432);        // 2*1024*64
  float* u     = ws + 917504;                // 2*128*1024
  float* v     = ws + 1179648;               // 2*128*1024
  float* pre   = ws + 1441792;               // 2*128*1024
  float* part  = ws + 1703936;               // 256 * 2*128
  float* stats = ws + 1769472;               // 2*128

  const float* xin = x0in;
  long xbs = (long)256 * N;                  // layer-0 input batch stride
  for (int l = 0; l < 4; ++l) {
    const int C = Cs[l], O = Os[l];
    knn_kernel<<<dim3(32, B), 256, 0, stream>>>(xin, xbs, C, idx);
    uv_gemm_kernel<<<dim3(N / 512, O / 16, B), 256, 0, stream>>>(W[l], xin, u, v, C, O, xbs);
    gather_stats_kernel<<<dim3(N / 8, B), O, 0, stream>>>(u, v, idx, pre, part, O);
    stats_reduce_kernel<<<1, 256, 0, stream>>>(part, stats, O, 2 * (N / 8));
    float* outp = xcon + (long)coff[l] * N;
    bn_act_kernel<<<(B * O * N) / 256, 256, 0, stream>>>(
        pre, stats, G[l], Bi[l], outp, O, (long)384 * N,
        1.0f / ((float)B * N * 64));
    xin = outp;                              // next layer reads its slice of xcon
    xbs = (long)384 * N;
  }
  gemm_kernel<<<dim3(N / 512, 384 / 16, B), 256, 0, stream>>>(
      wlast, xcon, (float*)d_out, 384, (long)384 * N, (long)384 * N);
}